// ChebConv_64390149701661
// MI455X (gfx1250) — compile-verified
//
#include <hip/hip_runtime.h>

typedef __attribute__((ext_vector_type(2))) float v2f;
typedef __attribute__((ext_vector_type(8))) float v8f;

#define CIN   128
#define COUT  128
#define BATCH 4
#define ROWF  512       // B*CIN floats per node row
#define KPSTRIDE 288    // floats per K-pair in LDS: 256 data + 32 pad (bank swizzle)

// ---------------------------------------------------------------------------
// x (B, Cin, V) row-major == (512, V) matrix  ->  x0 (V, 512)
// classic LDS-tiled transpose, 32x32 tiles, 256 threads
// ---------------------------------------------------------------------------
__global__ __launch_bounds__(256)
void transpose_in_kernel(const float* __restrict__ in, float* __restrict__ out, int V)
{
    __shared__ float tile[32][33];
    const int v0 = blockIdx.x * 32;
    const int r0 = blockIdx.y * 32;          // bc = b*Cin + c
    const int tx = threadIdx.x & 31;
    const int ty = threadIdx.x >> 5;         // 0..7
#pragma unroll
    for (int i = 0; i < 4; ++i) {
        int r = r0 + ty + i * 8;
        int v = v0 + tx;
        tile[ty + i * 8][tx] = (v < V) ? in[(long)r * V + v] : 0.0f;
    }
    __syncthreads();
#pragma unroll
    for (int i = 0; i < 4; ++i) {
        int v = v0 + ty + i * 8;
        if (v < V) out[(long)v * ROWF + r0 + tx] = tile[tx][ty + i * 8];
    }
}

// ---------------------------------------------------------------------------
// dst = 0
// ---------------------------------------------------------------------------
__global__ __launch_bounds__(256)
void zero_kernel(float4* __restrict__ p, long n4)
{
    long i = (long)blockIdx.x * blockDim.x + threadIdx.x;
    if (i < n4) p[i] = make_float4(0.f, 0.f, 0.f, 0.f);
}

// ---------------------------------------------------------------------------
// dst = -src  (init for x2 = 2*L*x1 - x0)
// ---------------------------------------------------------------------------
__global__ __launch_bounds__(256)
void negcopy_kernel(const float4* __restrict__ src, float4* __restrict__ dst, long n4)
{
    long i = (long)blockIdx.x * blockDim.x + threadIdx.x;
    if (i < n4) {
        float4 v = src[i];
        dst[i] = make_float4(-v.x, -v.y, -v.z, -v.w);
    }
}

// ---------------------------------------------------------------------------
// SpMM scatter: dst[rows[e], :] += alpha * vals[e] * src[cols[e], :]
// 2 edges per 256-thread block; each thread handles a float4 of the 512-row.
// Emits global_atomic_add_f32 (adds must stay f32 to match the reference).
// ---------------------------------------------------------------------------
__global__ __launch_bounds__(256)
void spmm_kernel(const float* __restrict__ src, float* __restrict__ dst,
                 const float* __restrict__ vals, const int* __restrict__ rows,
                 const int* __restrict__ cols, int E, float alpha)
{
    int e = blockIdx.x * 2 + (threadIdx.x >> 7);
    if (e >= E) return;
    const float a = alpha * vals[e];
    const long  r = rows[e];
    const long  c = cols[e];
    const int   f = (threadIdx.x & 127) * 4;

    const float4 xv = *(const float4*)(src + c * ROWF + f);
    float* d = dst + r * ROWF + f;
    atomicAdd(d + 0, a * xv.x);
    atomicAdd(d + 1, a * xv.y);
    atomicAdd(d + 2, a * xv.z);
    atomicAdd(d + 3, a * xv.w);
}

// ---------------------------------------------------------------------------
// Dense contraction via fp32 WMMA:
//   acc(M x 128) (+)= X(M x 128) @ W(128 x 128)          M = V*BATCH
// 256 threads = 8 waves; block tile = 128 rows x 128 cols, K = 128.
//
// W staged into LDS K-pair interleaved (sW[(k>>1)*KPSTRIDE + n*2 + (k&1)])
// -> one ds_load_b64 per B fragment (compiler fuses pairs to ds_load_2addr_b64),
// KPSTRIDE=288 floats keeps the two half-wave kpairs on disjoint bank halves.
//
// C preload/writeback: wave-uniform full-tile fast path (branch-free clause of
// base+imm global b32 ops); masked path only for the final partial M-tile.
// ---------------------------------------------------------------------------
__global__ __launch_bounds__(256, 2)
void cheb_gemm_kernel(const float* __restrict__ X, const float* __restrict__ W,
                      float* __restrict__ acc, int M, int accumulate)
{
    __shared__ float sW[(CIN / 2) * KPSTRIDE];   // 72 KB
    const int tid = threadIdx.x;

    // cooperative staging with K-pair interleave (coalesced float4 global reads)
#pragma unroll
    for (int it = 0; it < (CIN * COUT) / (256 * 4); ++it) {
        int idx = (it * 256 + tid) * 4;
        int r = idx >> 7;                        // K row
        int o = idx & 127;                       // N col
        float4 w = *(const float4*)(W + idx);
        float* d = sW + (r >> 1) * KPSTRIDE + (r & 1) + o * 2;
        d[0] = w.x; d[2] = w.y; d[4] = w.z; d[6] = w.w;
    }
    __syncthreads();

    const int lane  = tid & 31;
    const int wv    = tid >> 5;                  // wave 0..7
    const long r0   = (long)blockIdx.x * 128 + wv * 16;

    const int laneM = lane & 15;                 // M (A) / N (B,C) sub-index
    const int hi    = lane >> 4;                 // half-wave select
    const int khalf = hi * 2;                    // K offset: 0 or 2
    const int mhalf = hi * 8;                    // C-row offset: 0 or 8

    const bool full = (r0 + 16 <= (long)M);      // wave-uniform bounds test

    v8f c[8] = {};
    if (accumulate) {
        if (full) {
            const float* ap = acc + (r0 + mhalf) * COUT + laneM;
#pragma unroll
            for (int nt = 0; nt < 8; ++nt)
#pragma unroll
                for (int r = 0; r < 8; ++r)
                    c[nt][r] = ap[r * COUT + nt * 16];
        } else {
#pragma unroll
            for (int nt = 0; nt < 8; ++nt) {
                const long col = nt * 16 + laneM;
#pragma unroll
                for (int r = 0; r < 8; ++r) {
                    long row  = r0 + r + mhalf;
                    long rowc = (row < M) ? row : (M - 1);
                    c[nt][r] = acc[rowc * COUT + col];
                }
            }
        }
    }

    long arow = r0 + laneM;
    if (arow >= M) arow = M - 1;                 // clamp loads; stores are masked
    const float* Xr = X + arow * CIN + khalf;
    const float* Bp = sW + hi * KPSTRIDE + laneM * 2;
    __builtin_prefetch(Xr, 0, 3);                // global_prefetch_b8

#pragma unroll 2
    for (int kk = 0; kk < CIN / 4; ++kk) {
        v2f a = *(const v2f*)(Xr + kk * 4);      // global b64, 8B aligned
        const float* Bk = Bp + kk * (2 * KPSTRIDE);
        v2f b[8];
#pragma unroll
        for (int nt = 0; nt < 8; ++nt)           // ds_load_2addr_b64 pairs
            b[nt] = *(const v2f*)(Bk + nt * 32);
#pragma unroll
        for (int nt = 0; nt < 8; ++nt)
            c[nt] = __builtin_amdgcn_wmma_f32_16x16x4_f32(
                false, a, false, b[nt], (short)0, c[nt], false, false);
    }

    if (full) {
        float* ap = acc + (r0 + mhalf) * COUT + laneM;
#pragma unroll
        for (int nt = 0; nt < 8; ++nt)
#pragma unroll
            for (int r = 0; r < 8; ++r)
                ap[r * COUT + nt * 16] = c[nt][r];
    } else {
#pragma unroll
        for (int nt = 0; nt < 8; ++nt) {
            const long col = nt * 16 + laneM;
#pragma unroll
            for (int r = 0; r < 8; ++r) {
                long row = r0 + r + mhalf;
                if (row < M) acc[row * COUT + col] = c[nt][r];
            }
        }
    }
}

// ---------------------------------------------------------------------------
// acc (V,B,Cout) [row m = v*4+b]  ->  out (B, Cout, V), + bias, LDS-tiled
// ---------------------------------------------------------------------------
__global__ __launch_bounds__(256)
void epilogue_kernel(const float* __restrict__ acc, const float* __restrict__ bias,
                     float* __restrict__ out, int V)
{
    __shared__ float tile[32][33];
    const int v0 = blockIdx.x * 32;
    const int n0 = blockIdx.y * 32;
    const int b  = blockIdx.z;
    const int tx = threadIdx.x & 31;
    const int ty = threadIdx.x >> 5;
#pragma unroll
    for (int i = 0; i < 4; ++i) {
        int v = v0 + ty + i * 8;
        long m = (long)v * BATCH + b;
        tile[ty + i * 8][tx] = (v < V) ? acc[m * COUT + n0 + tx] : 0.0f;
    }
    __syncthreads();
#pragma unroll
    for (int i = 0; i < 4; ++i) {
        int n = n0 + ty + i * 8;
        int v = v0 + tx;
        if (v < V)
            out[((long)b * COUT + n) * V + v] = tile[tx][ty + i * 8] + bias[n];
    }
}

// ---------------------------------------------------------------------------
extern "C" void kernel_launch(void* const* d_in, const int* in_sizes, int n_in,
                              void* d_out, int out_size, void* d_ws, size_t ws_size,
                              hipStream_t stream)
{
    const float* x      = (const float*)d_in[0];
    const float* vals   = (const float*)d_in[1];
    const float* weight = (const float*)d_in[2];
    const float* bias   = (const float*)d_in[3];
    const int*   rows   = (const int*)d_in[4];
    const int*   cols   = (const int*)d_in[5];
    float* out = (float*)d_out;

    const int  V = in_sizes[0] / (BATCH * CIN);
    const int  E = in_sizes[1];
    const int  K = in_sizes[2] / (CIN * COUT);
    const int  M = V * BATCH;
    const long S = (long)V * ROWF;               // floats per Chebyshev state

    float* x0  = (float*)d_ws;
    float* x1  = x0 + S;
    float* x2  = x1 + S;
    float* acc = x2 + S;

    const dim3 blk(256);
    const int  vtiles = (V + 31) / 32;
    const int  mtiles = (M + 127) / 128;
    const long n4     = S / 4;
    const int  eb     = (int)((n4 + 255) / 256);
    const int  sb     = (E + 1) / 2;

    // x -> x0 (V, 512)
    transpose_in_kernel<<<dim3(vtiles, ROWF / 32), blk, 0, stream>>>(x, x0, V);

    // k = 0: acc = x0 @ W0
    cheb_gemm_kernel<<<mtiles, blk, 0, stream>>>(x0, weight, acc, M, 0);

    if (K > 1) {
        // x1 = L @ x0 ; acc += x1 @ W1
        zero_kernel<<<eb, blk, 0, stream>>>((float4*)x1, n4);
        spmm_kernel<<<sb, blk, 0, stream>>>(x0, x1, vals, rows, cols, E, 1.0f);
        cheb_gemm_kernel<<<mtiles, blk, 0, stream>>>(x1, weight + (long)CIN * COUT,
                                                     acc, M, 1);
        // k = 2..K-1: x2 = 2*L@x1 - x0 ; acc += x2 @ Wk ; rotate
        float *xa = x0, *xb = x1, *xc = x2;
        for (int k = 2; k < K; ++k) {
            negcopy_kernel<<<eb, blk, 0, stream>>>((const float4*)xa, (float4*)xc, n4);
            spmm_kernel<<<sb, blk, 0, stream>>>(xb, xc, vals, rows, cols, E, 2.0f);
            cheb_gemm_kernel<<<mtiles, blk, 0, stream>>>(
                xc, weight + (long)k * CIN * COUT, acc, M, 1);
            float* t = xa; xa = xb; xb = xc; xc = t;
        }
    }

    // acc -> out (B, Cout, V) + bias
    epilogue_kernel<<<dim3(vtiles, COUT / 32, BATCH), blk, 0, stream>>>(acc, bias, out, V);
}